// SOM_9062380995000
// MI455X (gfx1250) — compile-verified
//
#include <hip/hip_runtime.h>
#include <math.h>

// Problem constants (match reference)
#define B_ROWS   2048
#define DIMS     256
#define N_NODES  65536
#define TOPK     8

#define M_TILE   32          // query rows per workgroup (2 WMMA row-tiles)
#define WAVES    8           // wave32 waves per workgroup (256 threads)
#define AS_STRIDE 260        // padded LDS row stride for A tile (conflict-free b64 reads)
#define ST_STRIDE 17         // padded LDS row stride for distance staging
#define MAX_CHUNKS 32        // N split across gridDim.y

#define Y2_BYTES ((size_t)N_NODES * sizeof(float))

typedef __attribute__((ext_vector_type(2))) float v2f;
typedef __attribute__((ext_vector_type(8))) float v8f;

struct Cand { float v; int i; };

// Insert (d, idx) into an ascending sorted 8-list held in registers.
__device__ __forceinline__ void insert8(float (&bv)[TOPK], int (&bi)[TOPK], float d, int idx) {
  if (d >= bv[TOPK - 1]) return;
  #pragma unroll
  for (int j = TOPK - 1; j > 0; --j) {
    if (d < bv[j - 1]) { bv[j] = bv[j - 1]; bi[j] = bi[j - 1]; }
    else               { bv[j] = d;          bi[j] = idx; return; }
  }
  bv[0] = d; bi[0] = idx;
}

// Stage 0: node squared norms, computed once (removes per-k VALU work from hot loop).
__global__ __launch_bounds__(256) void som_node_norms(
    const float* __restrict__ nodes, float* __restrict__ y2) {
  const int n = blockIdx.x * 256 + threadIdx.x;
  const float4* p = (const float4*)(nodes + (size_t)n * DIMS);
  float s = 0.f;
  #pragma unroll 4
  for (int i = 0; i < DIMS / 4; ++i) {
    const float4 v = p[i];
    s = fmaf(v.x, v.x, s); s = fmaf(v.y, v.y, s);
    s = fmaf(v.z, v.z, s); s = fmaf(v.w, v.w, s);
  }
  y2[n] = s;
}

// Stage 1: fused fp32-WMMA distance GEMM (32xN tile) + per-chunk top-8.
__global__ __launch_bounds__(256) void som_dist_topk_stage1(
    const float* __restrict__ samples, const float* __restrict__ nodes,
    const float* __restrict__ y2g, Cand* __restrict__ ws,
    int chunkCols, int chunks) {
  __shared__ float As[M_TILE * AS_STRIDE];
  __shared__ float x2s[M_TILE];
  __shared__ float stage[WAVES * M_TILE * ST_STRIDE];
  __shared__ Cand lists[WAVES * M_TILE * TOPK];

  const int tid  = threadIdx.x;
  const int wave = tid >> 5;
  const int lane = tid & 31;
  const int half = lane >> 4;    // lane half selects k-pair (WMMA f32 16x16x4 layout)
  const int ln   = lane & 15;    // M (for A) / N (for B) index within tile

  const int rowBase   = blockIdx.x * M_TILE;
  const int chunkBase = blockIdx.y * chunkCols;

  // Cooperative load of the 32x256 A tile into LDS.
  for (int i = tid; i < M_TILE * DIMS; i += 256) {
    const int r = i >> 8, c = i & (DIMS - 1);
    As[r * AS_STRIDE + c] = samples[(size_t)(rowBase + r) * DIMS + c];
  }
  __syncthreads();
  if (tid < M_TILE) {
    float s = 0.f;
    for (int c = 0; c < DIMS; ++c) { const float a = As[tid * AS_STRIDE + c]; s = fmaf(a, a, s); }
    x2s[tid] = s;
  }
  __syncthreads();

  float best[TOPK]; int bidx[TOPK];
  #pragma unroll
  for (int j = 0; j < TOPK; ++j) { best[j] = 3.402823466e38f; bidx[j] = 0; }

  const int iters = chunkCols / (WAVES * 16);
  for (int it = 0; it < iters; ++it) {
    const int colBase = chunkBase + it * (WAVES * 16) + wave * 16;
    // B operand: column (node) ln of this 16-col tile; k offset 2*half per layout.
    const float* bp  = nodes + (size_t)(colBase + ln) * DIMS + 2 * half;
    const float* ap0 = &As[ln * AS_STRIDE + 2 * half];          // rows 0..15
    const float* ap1 = &As[(16 + ln) * AS_STRIDE + 2 * half];   // rows 16..31

    v8f acc0 = {0.f, 0.f, 0.f, 0.f, 0.f, 0.f, 0.f, 0.f};
    v8f acc1 = {0.f, 0.f, 0.f, 0.f, 0.f, 0.f, 0.f, 0.f};
    #pragma unroll 8
    for (int k0 = 0; k0 < DIMS; k0 += 4) {
      const v2f b  = *(const v2f*)(bp  + k0);   // global_load_b64 (L2-resident)
      const v2f a0 = *(const v2f*)(ap0 + k0);   // ds_load_b64
      const v2f a1 = *(const v2f*)(ap1 + k0);   // ds_load_b64 (B reused across 2 WMMAs)
      acc0 = __builtin_amdgcn_wmma_f32_16x16x4_f32(false, a0, false, b,
                                                   (short)0, acc0, false, false);
      acc1 = __builtin_amdgcn_wmma_f32_16x16x4_f32(false, a1, false, b,
                                                   (short)0, acc1, false, false);
    }
    const float y2v = y2g[colBase + ln];        // precomputed node norm

    // Distances -> wave-private LDS staging (C layout: row = v + 8*half [+16], col = ln).
    float* st = &stage[wave * M_TILE * ST_STRIDE];
    #pragma unroll
    for (int v = 0; v < 8; ++v) {
      const int m = v + 8 * half;
      const float sq0 = x2s[m]      + y2v - 2.0f * acc0[v];
      const float sq1 = x2s[m + 16] + y2v - 2.0f * acc1[v];
      st[m * ST_STRIDE + ln]        = sqrtf(fmaxf(sq0, 0.f));
      st[(m + 16) * ST_STRIDE + ln] = sqrtf(fmaxf(sq1, 0.f));
    }
    __syncthreads();
    // Row-per-lane remap: each of the 32 lanes owns one row, scans the 16 columns.
    {
      const float* rowp = &st[lane * ST_STRIDE];
      #pragma unroll
      for (int c = 0; c < 16; ++c) insert8(best, bidx, rowp[c], colBase + c);
    }
    __syncthreads();
  }

  // Publish per-wave sorted lists (one per row), merge 8 lists/row -> 8 cands/row/chunk.
  {
    Cand* lp = &lists[(wave * M_TILE + lane) * TOPK];
    #pragma unroll
    for (int j = 0; j < TOPK; ++j) { lp[j].v = best[j]; lp[j].i = bidx[j]; }
  }
  __syncthreads();
  if (tid < M_TILE) {
    float mb[TOPK]; int mi[TOPK];
    #pragma unroll
    for (int j = 0; j < TOPK; ++j) { mb[j] = 3.402823466e38f; mi[j] = 0; }
    for (int w = 0; w < WAVES; ++w) {
      const Cand* lp = &lists[(w * M_TILE + tid) * TOPK];
      #pragma unroll
      for (int j = 0; j < TOPK; ++j) {
        const Cand c = lp[j];
        if (c.v >= mb[TOPK - 1]) break;   // source list is sorted ascending
        insert8(mb, mi, c.v, c.i);
      }
    }
    Cand* op = ws + ((size_t)(rowBase + tid) * chunks + blockIdx.y) * TOPK;
    #pragma unroll
    for (int j = 0; j < TOPK; ++j) { op[j].v = mb[j]; op[j].i = mi[j]; }
  }
}

// Stage 2: one thread per row merges chunks*8 sorted candidates -> final ascending top-8.
__global__ __launch_bounds__(256) void som_topk_stage2(
    const Cand* __restrict__ ws, int* __restrict__ outIdx, float* __restrict__ outVal,
    int chunks) {
  const int row = blockIdx.x * blockDim.x + threadIdx.x;
  if (row >= B_ROWS) return;
  float best[TOPK]; int bidx[TOPK];
  #pragma unroll
  for (int j = 0; j < TOPK; ++j) { best[j] = 3.402823466e38f; bidx[j] = 0; }
  const Cand* p = ws + (size_t)row * chunks * TOPK;
  for (int c = 0; c < chunks; ++c) {
    const Cand* q = p + c * TOPK;
    #pragma unroll
    for (int j = 0; j < TOPK; ++j) {
      const Cand cd = q[j];
      if (cd.v >= best[TOPK - 1]) break;  // each group of 8 is sorted ascending
      insert8(best, bidx, cd.v, cd.i);
    }
  }
  #pragma unroll
  for (int j = 0; j < TOPK; ++j) {
    outIdx[row * TOPK + j] = bidx[j];
    outVal[row * TOPK + j] = best[j];
  }
}

extern "C" void kernel_launch(void* const* d_in, const int* in_sizes, int n_in,
                              void* d_out, int out_size, void* d_ws, size_t ws_size,
                              hipStream_t stream) {
  const float* samples = (const float*)d_in[0];   // [2048, 256] f32
  const float* nodes   = (const float*)d_in[1];   // [65536, 256] f32
  (void)in_sizes; (void)n_in; (void)out_size;     // n == TOPK == 8 (compile-time)

  // Workspace: [ y2: N floats | candidates: B * chunks * 8 Cand ]
  int chunks = MAX_CHUNKS;
  while (chunks > 1 &&
         Y2_BYTES + (size_t)B_ROWS * chunks * TOPK * sizeof(Cand) > ws_size) chunks >>= 1;
  const int chunkCols = N_NODES / chunks;

  float* y2  = (float*)d_ws;
  Cand*  cws = (Cand*)((char*)d_ws + Y2_BYTES);

  som_node_norms<<<dim3(N_NODES / 256), 256, 0, stream>>>(nodes, y2);
  som_dist_topk_stage1<<<dim3(B_ROWS / M_TILE, chunks), 256, 0, stream>>>(
      samples, nodes, y2, cws, chunkCols, chunks);

  // Output tuple: (indices int32 [2048,8], values f32 [2048,8]) concatenated flat.
  int*   outIdx = (int*)d_out;
  float* outVal = (float*)d_out + (size_t)B_ROWS * TOPK;
  som_topk_stage2<<<dim3((B_ROWS + 255) / 256), 256, 0, stream>>>(cws, outIdx, outVal, chunks);
}